// CostVolumeGWC_72602127171749
// MI455X (gfx1250) — compile-verified
//
#include <hip/hip_runtime.h>

// Problem constants (from reference setup_inputs)
#define BB 4
#define CC 320
#define HH 64
#define WW 128
#define DD 48
#define GG 40
#define CGR 8           // channels per group = C / G
#define TPAD 48         // left zero-padding of target row (>= maxdisp-1)
#define TW (WW + TPAD)  // 176
#define DPAD 16         // dump rows on each side of the d range
#define DROWS (DD + 2 * DPAD)  // 80: rows for d in [-16, 64)

typedef __attribute__((ext_vector_type(2))) float v2f;
typedef __attribute__((ext_vector_type(8))) float v8f;

// One block per (b, g, h). 256 threads = 8 wave32.
// cost[b, g, d, h, w] = (1/8) * sum_c ref[b, g*8+c, h, w] * tgt[b, g*8+c, h, w-d]
__global__ __launch_bounds__(256) void gwc_volume_wmma_kernel(
    const float* __restrict__ ref, const float* __restrict__ tgt,
    float* __restrict__ out) {
  __shared__ float s_ref[CGR][WW];      // 4 KB
  __shared__ float s_tgt[CGR][TW];      // 5.5 KB, columns [0,TPAD) are zero
  __shared__ float s_out[DROWS][WW];    // 40 KB; rows [DPAD, DPAD+48) are live

  const int tid = threadIdx.x;
  const int blk = blockIdx.x;           // 0 .. B*G*H-1
  const int h   = blk % HH;
  const int bg  = blk / HH;             // b*G + g
  const int b   = bg / GG;
  const int g   = bg % GG;

  const size_t in_base = ((size_t)(b * CC + g * CGR) * HH + h) * (size_t)WW;

  // ---- Stage input slices in LDS (each HBM byte read exactly once; NT) ----
  for (int idx = tid; idx < CGR * WW; idx += 256) {       // 1024, uniform trip
    const int c = idx >> 7;             // / 128
    const int w = idx & (WW - 1);
    const size_t off = in_base + (size_t)c * (HH * WW) + w;
    s_ref[c][w]        = __builtin_nontemporal_load(ref + off);
    s_tgt[c][TPAD + w] = __builtin_nontemporal_load(tgt + off);
  }
  for (int idx = tid; idx < CGR * TPAD; idx += 256) {     // zero the pad
    s_tgt[idx / TPAD][idx % TPAD] = 0.0f;
  }
  __syncthreads();

  const int lane = tid & 31;
  const int wave = tid >> 5;
  const int half = lane >> 4;           // 0: lanes 0-15, 1: lanes 16-31
  const int l16  = lane & 15;

  // ---- 32 tiles: w-tile a in [0,8), d-tile dt in [0,4); 4 tiles per wave ----
  for (int tt = 0; tt < 4; ++tt) {
    const int t  = wave * 4 + tt;
    const int a  = t >> 2;              // w tile
    const int dt = t & 3;               // d tile
    const int w0 = a * 16;
    const int d0 = dt * 16;
    // Column index into padded target row for B[k][j], j = l16:
    const int tcol = TPAD + w0 + l16 - d0;   // always >= 0

    v8f acc = {0.f, 0.f, 0.f, 0.f, 0.f, 0.f, 0.f, 0.f};

    // K = 8 channels as two K=4 f32 WMMAs.
    // A layout (16x4 f32): lane -> M = l16; VGPR0 = K(2*half), VGPR1 = K(2*half+1)
    // B layout (4x16 f32): lane -> N = l16; VGPR0 = K(2*half), VGPR1 = K(2*half+1)
    {
      const int c0 = 2 * half;          // channels 0..3
      v2f A, Bm;
      A.x  = s_ref[c0 + 0][w0 + l16];
      A.y  = s_ref[c0 + 1][w0 + l16];
      Bm.x = s_tgt[c0 + 0][tcol];
      Bm.y = s_tgt[c0 + 1][tcol];
      acc = __builtin_amdgcn_wmma_f32_16x16x4_f32(false, A, false, Bm,
                                                  (short)0, acc, false, false);
    }
    {
      const int c0 = 4 + 2 * half;      // channels 4..7
      v2f A, Bm;
      A.x  = s_ref[c0 + 0][w0 + l16];
      A.y  = s_ref[c0 + 1][w0 + l16];
      Bm.x = s_tgt[c0 + 0][tcol];
      Bm.y = s_tgt[c0 + 1][tcol];
      acc = __builtin_amdgcn_wmma_f32_16x16x4_f32(false, A, false, Bm,
                                                  (short)0, acc, false, false);
    }

    // C/D layout (16x16 f32, 8 VGPRs): lane -> N = l16, VGPR r -> M = r + 8*half.
    // Tile entry (M, N) -> output w = w0 + M, d = d0 + M - N, with
    // d in [-15, 63] always. Store UNCONDITIONALLY at row d + DPAD: rows
    // outside [DPAD, DPAD+48) are dump rows, each (row, w) is written at
    // most once across all tiles (unique-tile coverage), so no races.
#pragma unroll
    for (int r = 0; r < 8; ++r) {
      const int M   = r + 8 * half;
      const int w   = w0 + M;
      const int row = d0 + M - l16 + DPAD;   // in [1, 79]
      s_out[row][w] = acc[r] * 0.125f;
    }
  }
  __syncthreads();

  // ---- Coalesced non-temporal global store: [B,G,D,H,W] ----
  const size_t out_base = ((size_t)bg * DD) * (size_t)(HH * WW) + (size_t)h * WW;
  for (int idx = tid; idx < DD * WW; idx += 256) {        // 6144, uniform trip
    const int d = idx >> 7;
    const int w = idx & (WW - 1);
    __builtin_nontemporal_store(s_out[DPAD + d][w],
                                out + out_base + (size_t)d * (HH * WW) + w);
  }
}

extern "C" void kernel_launch(void* const* d_in, const int* in_sizes, int n_in,
                              void* d_out, int out_size, void* d_ws, size_t ws_size,
                              hipStream_t stream) {
  (void)in_sizes; (void)n_in; (void)d_ws; (void)ws_size; (void)out_size;
  const float* ref = (const float*)d_in[0];
  const float* tgt = (const float*)d_in[1];
  // d_in[2] = maxdisp (48), d_in[3] = num_group (40): fixed by the reference setup.
  float* out = (float*)d_out;
  gwc_volume_wmma_kernel<<<dim3(BB * GG * HH), dim3(256), 0, stream>>>(ref, tgt, out);
}